// GCNEncoder_64974265254502
// MI455X (gfx1250) — compile-verified
//
#include <hip/hip_runtime.h>
#include <hip/hip_bf16.h>
#include <stdint.h>

// ---------------------------------------------------------------------------
// Problem constants (match reference)
// ---------------------------------------------------------------------------
#define N_NODES 4096
#define E_EDGES 131072
#define INC     512
#define HID     512
#define OUTC    64
#define RANK    64
#define N_DOM   4
#define BN_EPS  1e-5f

typedef __attribute__((ext_vector_type(2))) float v2f;
typedef __attribute__((ext_vector_type(8))) float v8f;
typedef __attribute__((ext_vector_type(4))) int   v4i;

// ---------------------------------------------------------------------------
// CDNA5 async memory->LDS copy (ASYNCcnt-tracked) with portable fallback
//   Probe round 2: builtin exists; param0 is an int4* in AS1 (diagnostic:
//   "__vector_size__(4*sizeof(int)) int __device__ *"). Param1 -> LDS (AS3).
// ---------------------------------------------------------------------------
#if defined(__has_builtin)
#  if __has_builtin(__builtin_amdgcn_global_load_async_to_lds_b128) && \
      __has_builtin(__builtin_amdgcn_s_wait_asynccnt)
#    define USE_ASYNC_LDS 1
#  endif
#endif
#ifndef USE_ASYNC_LDS
#  define USE_ASYNC_LDS 0
#endif

#if USE_ASYNC_LDS
typedef __attribute__((address_space(1))) v4i g_v4i;
typedef __attribute__((address_space(3))) v4i l_v4i;
#endif

__device__ __forceinline__ void copy16_to_lds(const float* g, float* l) {
#if USE_ASYNC_LDS
  __builtin_amdgcn_global_load_async_to_lds_b128((g_v4i*)g, (l_v4i*)l, 0, 0);
#else
  *(float4*)l = *(const float4*)g;
#endif
}

__device__ __forceinline__ void stage_fence() {
#if USE_ASYNC_LDS
  __builtin_amdgcn_s_wait_asynccnt(0);   // own wave's async copies complete
#endif
  __syncthreads();                       // make LDS visible to all 4 waves
}

// ---------------------------------------------------------------------------
// FP32 WMMA GEMM, LDS-staged, double-buffered.
//   C[M x Nn] = op(A) @ B ; op(A) = A (row-major [M x K]) or A^T (stored [K x M])
// Block = 4 waves (32,4) -> one 64x64 C macro-tile.
//   wave w owns rows [64*bx + 16w, +16) and all 64 cols: 4 v8f accumulators,
//   so each A fragment feeds 4 V_WMMA_F32_16X16X4_F32 (32 WMMA per K-tile).
// LDS panels (per buffer): A 64x32 (layout [m][36] or [k][72]), B 32x64 ([k][72]).
//   Pads chosen so half-wave fragment reads hit disjoint bank sets.
// ---------------------------------------------------------------------------
#define TILE_K 32

template <bool TRANSA>
__global__ __launch_bounds__(128) void gemm_wmma_f32(
    const float* __restrict__ A, int lda,
    const float* __restrict__ B, int ldb,
    float* __restrict__ C, int ldc,
    int M, int Nn, int K) {
  __shared__ float As[2][2304];   // 64*36 (=2304) or 32*72 (=2304)
  __shared__ float Bs[2][2304];   // 32*72

  const int lane = threadIdx.x;            // 0..31
  const int lo   = lane & 15;
  const int hi   = lane >> 4;              // 0 or 1
  const int w    = threadIdx.y;            // wave 0..3
  const int tid  = lane + (w << 5);        // 0..127
  const int row0 = blockIdx.x * 64;
  const int col0 = blockIdx.y * 64;
  const int mrow = (w << 4) + lo;          // row within 64-row panel

  v8f acc0 = {}, acc1 = {}, acc2 = {}, acc3 = {};

  const int ntiles = K / TILE_K;

  auto stage = [&](int t, int buf) {
    const int k0 = t * TILE_K;
#pragma unroll
    for (int r = 0; r < 4; ++r) {
      const int idx = tid + (r << 7);                 // 0..511 (16B chunks)
      // ---- A panel ----
      if (TRANSA) {                                   // stored [K x M] -> LDS [k][72]
        const int ak = idx >> 4, q = (idx & 15) << 2;
        copy16_to_lds(A + (size_t)(k0 + ak) * lda + row0 + q,
                      &As[buf][ak * 72 + q]);
      } else {                                        // stored [M x K] -> LDS [m][36]
        const int am = idx >> 3, q = (idx & 7) << 2;
        copy16_to_lds(A + (size_t)(row0 + am) * lda + k0 + q,
                      &As[buf][am * 36 + q]);
      }
      // ---- B panel ----  [K x Nn] -> LDS [k][72]
      {
        const int bk = idx >> 4, q = (idx & 15) << 2;
        copy16_to_lds(B + (size_t)(k0 + bk) * ldb + col0 + q,
                      &Bs[buf][bk * 72 + q]);
      }
    }
  };

  auto compute = [&](int buf) {
    const float* __restrict__ AsB = As[buf];
    const float* __restrict__ BsB = Bs[buf];
#pragma unroll
    for (int kk = 0; kk < TILE_K; kk += 4) {
      const int ka = kk + 2 * hi;
      v2f a;
      if (TRANSA) { a.x = AsB[ka * 72 + mrow];  a.y = AsB[(ka + 1) * 72 + mrow]; }
      else        { a.x = AsB[mrow * 36 + ka];  a.y = AsB[mrow * 36 + ka + 1];  }
      v2f b;
      b.x = BsB[ka * 72 +  0 + lo]; b.y = BsB[(ka + 1) * 72 +  0 + lo];
      acc0 = __builtin_amdgcn_wmma_f32_16x16x4_f32(false, a, false, b, (short)0, acc0, false, false);
      b.x = BsB[ka * 72 + 16 + lo]; b.y = BsB[(ka + 1) * 72 + 16 + lo];
      acc1 = __builtin_amdgcn_wmma_f32_16x16x4_f32(false, a, false, b, (short)0, acc1, false, false);
      b.x = BsB[ka * 72 + 32 + lo]; b.y = BsB[(ka + 1) * 72 + 32 + lo];
      acc2 = __builtin_amdgcn_wmma_f32_16x16x4_f32(false, a, false, b, (short)0, acc2, false, false);
      b.x = BsB[ka * 72 + 48 + lo]; b.y = BsB[(ka + 1) * 72 + 48 + lo];
      acc3 = __builtin_amdgcn_wmma_f32_16x16x4_f32(false, a, false, b, (short)0, acc3, false, false);
    }
  };

  stage(0, 0);
  stage_fence();
  for (int t = 0; t < ntiles; ++t) {
    const int buf = t & 1;
    if (t + 1 < ntiles) stage(t + 1, buf ^ 1);  // prefetch next panels
    compute(buf);
    stage_fence();                              // copies done + all waves past reads
  }

#pragma unroll
  for (int r = 0; r < 8; ++r) {
    const int rr = row0 + (w << 4) + r + 8 * hi;
    C[(size_t)rr * ldc + col0 +  0 + lo] = acc0[r];
    C[(size_t)rr * ldc + col0 + 16 + lo] = acc1[r];
    C[(size_t)rr * ldc + col0 + 32 + lo] = acc2[r];
    C[(size_t)rr * ldc + col0 + 48 + lo] = acc3[r];
  }
}

// ---------------------------------------------------------------------------
// Small elementwise / graph kernels
// ---------------------------------------------------------------------------
__global__ void fill_kernel(float* p, float v, int n) {
  int i = blockIdx.x * blockDim.x + threadIdx.x;
  if (i < n) p[i] = v;
}

__global__ void edge_deg_kernel(const int* __restrict__ col,
                                const float* __restrict__ w,
                                float* __restrict__ deg) {
  int e = blockIdx.x * blockDim.x + threadIdx.x;
  if (e < E_EDGES) atomicAdd(&deg[col[e]], w[e]);
}

__global__ void safe_rsqrt_kernel(const float* __restrict__ deg,
                                  float* __restrict__ dinv, int n) {
  int i = blockIdx.x * blockDim.x + threadIdx.x;
  if (i >= n) return;
  float d = deg[i];
  dinv[i] = (d > 0.0f) ? rsqrtf(d) : 0.0f;
}

// out[i,c] = dinv[i]^2 * Xh[i,c] + bias[c]   (self-loop term)
__global__ void agg_init_kernel(const float* __restrict__ dinv,
                                const float* __restrict__ Xh,
                                const float* __restrict__ bias,
                                float* __restrict__ out, int C) {
  int idx = blockIdx.x * blockDim.x + threadIdx.x;
  if (idx >= N_NODES * C) return;
  int i = idx / C, ch = idx - i * C;
  float d = dinv[i];
  out[idx] = d * d * Xh[idx] + bias[ch];
}

// out[col,c] += dinv[row]*w*dinv[col] * Xh[row,c] ; one wave per edge
__global__ void agg_edges_kernel(const int* __restrict__ row,
                                 const int* __restrict__ col,
                                 const float* __restrict__ w,
                                 const float* __restrict__ dinv,
                                 const float* __restrict__ Xh,
                                 float* __restrict__ out, int C) {
  int g = blockIdx.x * blockDim.x + threadIdx.x;
  int e = g >> 5, lane = g & 31;
  if (e >= E_EDGES) return;
  int r = row[e], c = col[e];
  float coef = dinv[r] * w[e] * dinv[c];
  for (int ch = lane; ch < C; ch += 32)
    atomicAdd(&out[c * C + ch], coef * Xh[r * C + ch]);
}

// Y[dst[e],k] += w[e] * X[src[e],k] ; one wave per edge, RANK=64 cols
__global__ void spmm_rank_kernel(const int* __restrict__ dst,
                                 const int* __restrict__ src,
                                 const float* __restrict__ w,
                                 const float* __restrict__ X,
                                 float* __restrict__ Y) {
  int g = blockIdx.x * blockDim.x + threadIdx.x;
  int e = g >> 5, lane = g & 31;
  if (e >= E_EDGES) return;
  int d = dst[e], s = src[e];
  float we = w[e];
  atomicAdd(&Y[d * RANK + lane],      we * X[s * RANK + lane]);
  atomicAdd(&Y[d * RANK + lane + 32], we * X[s * RANK + lane + 32]);
}

// ---------------------------------------------------------------------------
// DSBN (per-domain training-mode BatchNorm) + ReLU
// ---------------------------------------------------------------------------
__global__ void bn_count_kernel(const int* __restrict__ y, int* __restrict__ cnt) {
  int i = blockIdx.x * blockDim.x + threadIdx.x;
  if (i < N_NODES) atomicAdd(&cnt[y[i]], 1);
}

__global__ void bn_accum_kernel(const float* __restrict__ xpre,
                                const int* __restrict__ y,
                                float* __restrict__ sum, float* __restrict__ sq) {
  int idx = blockIdx.x * blockDim.x + threadIdx.x;
  if (idx >= N_NODES * HID) return;
  int i = idx / HID, ch = idx - i * HID;
  float v = xpre[idx];
  int d = y[i];
  atomicAdd(&sum[d * HID + ch], v);
  atomicAdd(&sq[d * HID + ch], v * v);
}

__global__ void bn_final_kernel(const float* __restrict__ sum,
                                const float* __restrict__ sq,
                                const int* __restrict__ cnt,
                                const float* __restrict__ gamma,
                                const float* __restrict__ beta,
                                float* __restrict__ scale,
                                float* __restrict__ shift) {
  int idx = blockIdx.x * blockDim.x + threadIdx.x;
  if (idx >= N_DOM * HID) return;
  int d = idx / HID;
  float c    = fmaxf((float)cnt[d], 1.0f);
  float mean = sum[idx] / c;
  float var  = fmaxf(sq[idx] / c - mean * mean, 0.0f);
  float sc   = gamma[idx] * rsqrtf(var + BN_EPS);
  scale[idx] = sc;
  shift[idx] = beta[idx] - mean * sc;
}

__global__ void bn_apply_kernel(const float* __restrict__ xpre,
                                const int* __restrict__ y,
                                const float* __restrict__ scale,
                                const float* __restrict__ shift,
                                float* __restrict__ out) {
  int idx = blockIdx.x * blockDim.x + threadIdx.x;
  if (idx >= N_NODES * HID) return;
  int i = idx / HID, ch = idx - i * HID;
  int d = y[i];
  float v = xpre[idx] * scale[d * HID + ch] + shift[d * HID + ch];
  out[idx] = fmaxf(v, 0.0f);
}

// ---------------------------------------------------------------------------
// Randomized-subspace helpers
// ---------------------------------------------------------------------------
__device__ inline uint32_t hash_u32(uint32_t x) {
  x ^= x >> 17; x *= 0xed5ad4bbu; x ^= x >> 11; x *= 0xac4c1b51u;
  x ^= x >> 15; x *= 0x31848babu; x ^= x >> 14; return x;
}

__global__ void gen_omega_kernel(float* __restrict__ Om) {
  int idx = blockIdx.x * blockDim.x + threadIdx.x;
  if (idx >= N_NODES * RANK) return;
  uint32_t h1 = hash_u32(2u * (uint32_t)idx + 0x9e3779b9u);
  uint32_t h2 = hash_u32(2u * (uint32_t)idx + 0x7f4a7c15u);
  float u1 = ((float)h1 + 0.5f) * (1.0f / 4294967296.0f);
  float u2 = (float)h2 * (1.0f / 4294967296.0f);
  Om[idx] = sqrtf(-2.0f * logf(u1)) * cosf(6.28318530718f * u2);
}

// Cholesky of G (64x64, G = R^T R, upper R) + upper-triangular inverse.
__global__ void chol_inv64_kernel(const float* __restrict__ G,
                                  float* __restrict__ Rinv) {
  __shared__ float Gs[RANK * RANK];
  int t = threadIdx.x;
  for (int i = 0; i < RANK; ++i) Gs[i * RANK + t] = G[i * RANK + t];
  __syncthreads();
  for (int k = 0; k < RANK; ++k) {
    float dk = sqrtf(fmaxf(Gs[k * RANK + k], 1e-30f));
    __syncthreads();
    if (t == k)      Gs[k * RANK + k] = dk;
    else if (t > k)  Gs[k * RANK + t] /= dk;
    __syncthreads();
    float rkt = (t > k) ? Gs[k * RANK + t] : 0.0f;
    for (int i = k + 1; i < RANK; ++i)
      if (t >= i) Gs[i * RANK + t] -= Gs[k * RANK + i] * rkt;
    __syncthreads();
  }
  float xcol[RANK];
  for (int i = RANK - 1; i >= 0; --i) {
    if (i > t) { xcol[i] = 0.0f; continue; }
    float s = (i == t) ? 1.0f : 0.0f;
    for (int j = i + 1; j <= t; ++j) s -= Gs[i * RANK + j] * xcol[j];
    xcol[i] = s / Gs[i * RANK + i];
  }
  for (int i = 0; i < RANK; ++i)
    Rinv[i * RANK + t] = (i <= t) ? xcol[i] : 0.0f;
}

__global__ void colsum_rank_kernel(const float* __restrict__ Q,
                                   float* __restrict__ qsum) {
  int t = threadIdx.x;
  float s = 0.0f;
  for (int i = 0; i < N_NODES; ++i) s += Q[i * RANK + t];
  qsum[t] = s;
}

__global__ void deg_lowrank_kernel(const float* __restrict__ Rt,
                                   const float* __restrict__ qsum,
                                   float* __restrict__ deg) {
  int j = blockIdx.x * blockDim.x + threadIdx.x;
  if (j >= N_NODES) return;
  float s = 0.0f;
  for (int k = 0; k < RANK; ++k) s += Rt[j * RANK + k] * qsum[k];
  deg[j] = s;
}

__global__ void rowscale_kernel(const float* __restrict__ src,
                                const float* __restrict__ s,
                                float* __restrict__ dst, int C) {
  int idx = blockIdx.x * blockDim.x + threadIdx.x;
  if (idx >= N_NODES * C) return;
  int i = idx / C;
  dst[idx] = s[i] * src[idx];
}

__global__ void rowscale_bias_kernel(float* __restrict__ x,
                                     const float* __restrict__ s,
                                     const float* __restrict__ b, int C) {
  int idx = blockIdx.x * blockDim.x + threadIdx.x;
  if (idx >= N_NODES * C) return;
  int i = idx / C, ch = idx - i * C;
  x[idx] = s[i] * x[idx] + b[ch];
}

// ---------------------------------------------------------------------------
// Host orchestration
// ---------------------------------------------------------------------------
static inline void gemm_nn(const float* A, int lda, const float* B, int ldb,
                           float* C, int ldc, int M, int Nn, int K,
                           hipStream_t s) {
  dim3 g(M / 64, Nn / 64), b(32, 4);
  gemm_wmma_f32<false><<<g, b, 0, s>>>(A, lda, B, ldb, C, ldc, M, Nn, K);
}
static inline void gemm_tn(const float* A, int lda, const float* B, int ldb,
                           float* C, int ldc, int M, int Nn, int K,
                           hipStream_t s) {
  dim3 g(M / 64, Nn / 64), b(32, 4);
  gemm_wmma_f32<true><<<g, b, 0, s>>>(A, lda, B, ldb, C, ldc, M, Nn, K);
}

extern "C" void kernel_launch(void* const* d_in, const int* in_sizes, int n_in,
                              void* d_out, int out_size, void* d_ws, size_t ws_size,
                              hipStream_t stream) {
  const float* x     = (const float*)d_in[0];
  const int*   ei    = (const int*)d_in[1];
  const int*   row   = ei;
  const int*   col   = ei + E_EDGES;
  const float* w     = (const float*)d_in[2];
  const int*   y     = (const int*)d_in[3];
  const float* W1    = (const float*)d_in[4];
  const float* b1    = (const float*)d_in[5];
  const float* Wmu   = (const float*)d_in[6];
  const float* bmu   = (const float*)d_in[7];
  const float* Wlv   = (const float*)d_in[8];
  const float* blv   = (const float*)d_in[9];
  const float* gamma = (const float*)d_in[10];
  const float* beta  = (const float*)d_in[11];

  float* ws = (float*)d_ws;
  size_t off = 0;
  float* xw    = ws + off; off += (size_t)N_NODES * HID;
  float* pre   = ws + off; off += (size_t)N_NODES * HID;
  float* h     = ws + off; off += (size_t)N_NODES * HID;
  float* hw    = ws + off; off += (size_t)N_NODES * OUTC;
  float* Ybuf  = ws + off; off += (size_t)N_NODES * RANK;
  float* Q     = ws + off; off += (size_t)N_NODES * RANK;
  float* Rt    = ws + off; off += (size_t)N_NODES * RANK;
  float* T2    = ws + off; off += (size_t)RANK * HID;
  float* T2h   = ws + off; off += (size_t)RANK * OUTC;
  float* G     = ws + off; off += (size_t)RANK * RANK;
  float* Rinv  = ws + off; off += (size_t)RANK * RANK;
  float* deg1  = ws + off; off += N_NODES;
  float* dinv1 = ws + off; off += N_NODES;
  float* deg2  = ws + off; off += N_NODES;
  float* dinv2 = ws + off; off += N_NODES;
  float* qsum  = ws + off; off += RANK;
  float* bnSum = ws + off; off += (size_t)N_DOM * HID;
  float* bnSq  = ws + off; off += (size_t)N_DOM * HID;
  float* bnSc  = ws + off; off += (size_t)N_DOM * HID;
  float* bnSh  = ws + off; off += (size_t)N_DOM * HID;
  int*   bnCnt = (int*)(ws + off); off += N_DOM;
  if (ws_size < off * sizeof(float)) return;

  const int NM = N_NODES * OUTC;
  float* z1mu = (float*)d_out;
  float* z1lv = z1mu + NM;
  float* z2mu = z1mu + 2 * NM;
  float* z2lv = z1mu + 3 * NM;

  const int TB = 256;
  const int gNH = (N_NODES * HID) / TB;
  const int gNO = (N_NODES * OUTC) / TB;
  const int gN  = N_NODES / TB;
  const int gE  = E_EDGES / TB;
  const int gEW = (E_EDGES * 32) / TB;

  // ===================== shared: xw = x @ W1 ==============================
  gemm_nn(x, INC, W1, HID, xw, HID, N_NODES, HID, INC, stream);

  // ===================== branch 1: sparse graph ===========================
  fill_kernel<<<gN, TB, 0, stream>>>(deg1, 1.0f, N_NODES);
  edge_deg_kernel<<<gE, TB, 0, stream>>>(col, w, deg1);
  safe_rsqrt_kernel<<<gN, TB, 0, stream>>>(deg1, dinv1, N_NODES);

  agg_init_kernel<<<gNH, TB, 0, stream>>>(dinv1, xw, b1, pre, HID);
  agg_edges_kernel<<<gEW, TB, 0, stream>>>(row, col, w, dinv1, xw, pre, HID);

  (void)hipMemsetAsync(bnSum, 0, N_DOM * HID * sizeof(float), stream);
  (void)hipMemsetAsync(bnSq,  0, N_DOM * HID * sizeof(float), stream);
  (void)hipMemsetAsync(bnCnt, 0, N_DOM * sizeof(int), stream);
  bn_count_kernel<<<gN, TB, 0, stream>>>(y, bnCnt);
  bn_accum_kernel<<<gNH, TB, 0, stream>>>(pre, y, bnSum, bnSq);
  bn_final_kernel<<<(N_DOM * HID) / TB, TB, 0, stream>>>(bnSum, bnSq, bnCnt, gamma, beta, bnSc, bnSh);
  bn_apply_kernel<<<gNH, TB, 0, stream>>>(pre, y, bnSc, bnSh, h);  // h1

  gemm_nn(h, HID, Wmu, OUTC, hw, OUTC, N_NODES, OUTC, HID, stream);
  agg_init_kernel<<<gNO, TB, 0, stream>>>(dinv1, hw, bmu, z1mu, OUTC);
  agg_edges_kernel<<<gEW, TB, 0, stream>>>(row, col, w, dinv1, hw, z1mu, OUTC);

  gemm_nn(h, HID, Wlv, OUTC, hw, OUTC, N_NODES, OUTC, HID, stream);
  agg_init_kernel<<<gNO, TB, 0, stream>>>(dinv1, hw, blv, z1lv, OUTC);
  agg_edges_kernel<<<gEW, TB, 0, stream>>>(row, col, w, dinv1, hw, z1lv, OUTC);

  // ============== branch 2: randomized subspace (A2 = Q Q^T A) ============
  gen_omega_kernel<<<(N_NODES * RANK) / TB, TB, 0, stream>>>(Q);

  for (int step = 0; step < 5; ++step) {
    const int* dI = (step & 1) ? col : row;   // even: A @ Q ; odd: A^T @ Q
    const int* sI = (step & 1) ? row : col;
    (void)hipMemsetAsync(Ybuf, 0, (size_t)N_NODES * RANK * sizeof(float), stream);
    spmm_rank_kernel<<<gEW, TB, 0, stream>>>(dI, sI, w, Q, Ybuf);
    gemm_tn(Ybuf, RANK, Ybuf, RANK, G, RANK, RANK, RANK, N_NODES, stream);
    chol_inv64_kernel<<<1, RANK, 0, stream>>>(G, Rinv);
    gemm_nn(Ybuf, RANK, Rinv, RANK, Q, RANK, N_NODES, RANK, RANK, stream);
  }

  (void)hipMemsetAsync(Rt, 0, (size_t)N_NODES * RANK * sizeof(float), stream);
  spmm_rank_kernel<<<gEW, TB, 0, stream>>>(col, row, w, Q, Rt);

  colsum_rank_kernel<<<1, RANK, 0, stream>>>(Q, qsum);
  deg_lowrank_kernel<<<gN, TB, 0, stream>>>(Rt, qsum, deg2);
  safe_rsqrt_kernel<<<gN, TB, 0, stream>>>(deg2, dinv2, N_NODES);

  // h2pre = dinv2 .* (Rt @ (Q^T @ (dinv2 .* xw))) + b1
  rowscale_kernel<<<gNH, TB, 0, stream>>>(xw, dinv2, pre, HID);
  gemm_tn(Q, RANK, pre, HID, T2, HID, RANK, HID, N_NODES, stream);
  gemm_nn(Rt, RANK, T2, HID, pre, HID, N_NODES, HID, RANK, stream);
  rowscale_bias_kernel<<<gNH, TB, 0, stream>>>(pre, dinv2, b1, HID);

  (void)hipMemsetAsync(bnSum, 0, N_DOM * HID * sizeof(float), stream);
  (void)hipMemsetAsync(bnSq,  0, N_DOM * HID * sizeof(float), stream);
  (void)hipMemsetAsync(bnCnt, 0, N_DOM * sizeof(int), stream);
  bn_count_kernel<<<gN, TB, 0, stream>>>(y, bnCnt);
  bn_accum_kernel<<<gNH, TB, 0, stream>>>(pre, y, bnSum, bnSq);
  bn_final_kernel<<<(N_DOM * HID) / TB, TB, 0, stream>>>(bnSum, bnSq, bnCnt, gamma, beta, bnSc, bnSh);
  bn_apply_kernel<<<gNH, TB, 0, stream>>>(pre, y, bnSc, bnSh, h);  // h2

  gemm_nn(h, HID, Wmu, OUTC, hw, OUTC, N_NODES, OUTC, HID, stream);
  rowscale_kernel<<<gNO, TB, 0, stream>>>(hw, dinv2, hw, OUTC);
  gemm_tn(Q, RANK, hw, OUTC, T2h, OUTC, RANK, OUTC, N_NODES, stream);
  gemm_nn(Rt, RANK, T2h, OUTC, z2mu, OUTC, N_NODES, OUTC, RANK, stream);
  rowscale_bias_kernel<<<gNO, TB, 0, stream>>>(z2mu, dinv2, bmu, OUTC);

  gemm_nn(h, HID, Wlv, OUTC, hw, OUTC, N_NODES, OUTC, HID, stream);
  rowscale_kernel<<<gNO, TB, 0, stream>>>(hw, dinv2, hw, OUTC);
  gemm_tn(Q, RANK, hw, OUTC, T2h, OUTC, RANK, OUTC, N_NODES, stream);
  gemm_nn(Rt, RANK, T2h, OUTC, z2lv, OUTC, N_NODES, OUTC, RANK, stream);
  rowscale_bias_kernel<<<gNO, TB, 0, stream>>>(z2lv, dinv2, blv, OUTC);

  (void)in_sizes; (void)n_in; (void)out_size;
}